// TransEModel_73478300500125
// MI455X (gfx1250) — compile-verified
//
#include <hip/hip_runtime.h>
#include <stdint.h>

// Problem constants (match reference setup_inputs)
#define E_ENT 100000
#define R_REL 1000
#define D_DIM 200
#define B_BAT 256
#define BN_EPS 1e-5f

// Tiling
#define BT 64      // batch rows per workgroup tile
#define ET 64      // entities per workgroup tile
#define DK 8       // d-chunk per pipeline stage
#define NCHUNK (D_DIM / DK)   // 25

#if __has_builtin(__builtin_amdgcn_global_load_async_to_lds_b128)
#define HAVE_ASYNC 1
#endif

// Pointer types the async builtin expects: v4i in AS1 (global) / AS3 (LDS).
typedef int v4i_ __attribute__((ext_vector_type(4)));
typedef __attribute__((address_space(1))) v4i_* g_v4i_ptr;
typedef __attribute__((address_space(3))) v4i_* l_v4i_ptr;

// ---- CDNA5 async global -> LDS copy (16 bytes per lane) -------------------
// Generic->AS casts via integer round-trip (flat LDS addr low 32 bits == LDS
// byte offset per CDNA5 aperture rules).
__device__ __forceinline__ void copy16_g2l(void* lds, const void* g) {
#ifdef HAVE_ASYNC
  __builtin_amdgcn_global_load_async_to_lds_b128(
      (g_v4i_ptr)(unsigned long long)(g),
      (l_v4i_ptr)(unsigned int)(unsigned long long)(lds),
      /*offset=*/0, /*cpol=*/0);
#else
  *(float4*)lds = *(const float4*)g;
#endif
}

__device__ __forceinline__ void wait_async0() {
#ifdef HAVE_ASYNC
#if __has_builtin(__builtin_amdgcn_s_wait_asynccnt)
  __builtin_amdgcn_s_wait_asynccnt(0);
#else
  asm volatile("s_wait_asynccnt 0" ::: "memory");
#endif
#endif
}

// ---- Kernel: out[b][e] = -sum_d |x[b][d] - emb_e[e][d]| -------------------
// (placed first in the TU so the disasm snippet shows the async pipeline)
// Grid: (ceil(E/ET), B/BT). Block: 256 threads = 16(tx: entity) x 16(ty: batch).
// Per-thread register tile: 4 batch rows (b = ty*4+i) x 4 entities (e = tx+16*j).
__global__ __launch_bounds__(256) void transe_l1dist(
    const float* __restrict__ emb_e, const float* __restrict__ x,
    float* __restrict__ out) {
  __shared__ __align__(16) float xs[BT * D_DIM];      // 51200 B: full x tile
  __shared__ __align__(16) float es[2][ET * DK];      // 2 x 2048 B: emb chunks

  const int tid = threadIdx.x;
  const int tx = tid & 15;
  const int ty = tid >> 4;
  const int eBase = blockIdx.x * ET;
  const int bBase = blockIdx.y * BT;

  // --- preload full x tile (contiguous 51200 B) via async copies ---
  {
    const float* xsrc = x + (size_t)bBase * D_DIM;
    const int nvec = (BT * D_DIM) / 4;  // 3200 x b128
    for (int i = tid; i < nvec; i += 256)
      copy16_g2l(&xs[i * 4], xsrc + (size_t)i * 4);
  }

  // --- emb chunk loader: 64 rows x 32 B = 128 b128 transfers (threads 0..127)
  auto load_echunk = [&](int c, int buf) {
    if (tid < 128) {
      int row = tid >> 1;
      int half = tid & 1;                 // which 16 B of the 32 B row-chunk
      int er = eBase + row;
      if (er >= E_ENT) er = E_ENT - 1;    // clamp: loads stay in-bounds
      const float* g = emb_e + (size_t)er * D_DIM + c * DK + half * 4;
      copy16_g2l(&es[buf][row * DK + half * 4], g);
    }
  };

  load_echunk(0, 0);
  wait_async0();
  __syncthreads();

  float acc[4][4];
#pragma unroll
  for (int i = 0; i < 4; ++i)
#pragma unroll
    for (int j = 0; j < 4; ++j) acc[i][j] = 0.0f;

  for (int c = 0; c < NCHUNK; ++c) {
    const int cur = c & 1;
    if (c + 1 < NCHUNK) load_echunk(c + 1, cur ^ 1);  // overlap DMA w/ compute

    // entity fragments: e = tx + 16*j, 8 d-values each (two float4)
    float4 ev0[4], ev1[4];
#pragma unroll
    for (int j = 0; j < 4; ++j) {
      const int e = tx + 16 * j;
      ev0[j] = *(const float4*)&es[cur][e * DK];
      ev1[j] = *(const float4*)&es[cur][e * DK + 4];
    }

#pragma unroll
    for (int i = 0; i < 4; ++i) {
      const int b = ty * 4 + i;
      const float4 xv0 = *(const float4*)&xs[b * D_DIM + c * DK];
      const float4 xv1 = *(const float4*)&xs[b * D_DIM + c * DK + 4];
#pragma unroll
      for (int j = 0; j < 4; ++j) {
        // 8 subs + 8 adds (abs folds into VOP3 source modifiers)
        float s01 = fabsf(xv0.x - ev0[j].x) + fabsf(xv0.y - ev0[j].y);
        float s23 = fabsf(xv0.z - ev0[j].z) + fabsf(xv0.w - ev0[j].w);
        float s45 = fabsf(xv1.x - ev1[j].x) + fabsf(xv1.y - ev1[j].y);
        float s67 = fabsf(xv1.z - ev1[j].z) + fabsf(xv1.w - ev1[j].w);
        acc[i][j] += (s01 + s23) + (s45 + s67);
      }
    }

    wait_async0();     // next chunk's DMA has landed
    __syncthreads();   // everyone done reading 'cur' before it's overwritten
  }

  // --- store: out[b][e] = -acc, row-major [B, E] ---
  // Non-temporal: output is write-once/never-read; keep L2 for emb_e (80 MB
  // fits the 192 MB L2 and is re-read by all 4 batch tiles).
#pragma unroll
  for (int i = 0; i < 4; ++i) {
    const int bg = bBase + ty * 4 + i;
#pragma unroll
    for (int j = 0; j < 4; ++j) {
      const int eg = eBase + tx + 16 * j;
      if (eg < E_ENT)
        __builtin_nontemporal_store(-acc[i][j], &out[(size_t)bg * E_ENT + eg]);
    }
  }
}

// ---- Kernel: x[b][d] = BN0(emb_e[h_idx[b]][d]) + BN1(emb_r[r_idx[b]][d])
__global__ __launch_bounds__(256) void transe_prep(
    const int* __restrict__ h_idx, const int* __restrict__ r_idx,
    const float* __restrict__ emb_e, const float* __restrict__ emb_r,
    const float* __restrict__ g0, const float* __restrict__ b0,
    const float* __restrict__ m0, const float* __restrict__ v0,
    const float* __restrict__ g1, const float* __restrict__ b1,
    const float* __restrict__ m1, const float* __restrict__ v1,
    float* __restrict__ x) {
  int b = blockIdx.x;
  int d = threadIdx.x;
  if (d >= D_DIM) return;
  int he = h_idx[b];
  int re = r_idx[b];
  float ve = emb_e[(size_t)he * D_DIM + d];
  float vr = emb_r[(size_t)re * D_DIM + d];
  float h = g0[d] * (ve - m0[d]) * rsqrtf(v0[d] + BN_EPS) + b0[d];
  float r = g1[d] * (vr - m1[d]) * rsqrtf(v1[d] + BN_EPS) + b1[d];
  x[(size_t)b * D_DIM + d] = h + r;
}

extern "C" void kernel_launch(void* const* d_in, const int* in_sizes, int n_in,
                              void* d_out, int out_size, void* d_ws, size_t ws_size,
                              hipStream_t stream) {
  const int* h_idx = (const int*)d_in[0];
  const int* r_idx = (const int*)d_in[1];
  const float* emb_e = (const float*)d_in[2];
  const float* emb_r = (const float*)d_in[3];
  const float* g0 = (const float*)d_in[4];
  const float* b0 = (const float*)d_in[5];
  const float* m0 = (const float*)d_in[6];
  const float* v0 = (const float*)d_in[7];
  const float* g1 = (const float*)d_in[8];
  const float* b1 = (const float*)d_in[9];
  const float* m1 = (const float*)d_in[10];
  const float* v1 = (const float*)d_in[11];

  float* xbuf = (float*)d_ws;          // B*D floats = 204800 B of scratch
  float* out = (float*)d_out;          // [B, E] fp32

  transe_prep<<<dim3(B_BAT), dim3(256), 0, stream>>>(
      h_idx, r_idx, emb_e, emb_r, g0, b0, m0, v0, g1, b1, m1, v1, xbuf);

  dim3 grid((E_ENT + ET - 1) / ET, B_BAT / BT);  // (1563, 4)
  transe_l1dist<<<grid, dim3(256), 0, stream>>>(emb_e, xbuf, out);
}